// LightweightTransformerPatchAttention_55259049230700
// MI455X (gfx1250) — compile-verified
//
#include <hip/hip_runtime.h>
#include <cstdint>

typedef float v2f  __attribute__((ext_vector_type(2)));
typedef float v8f  __attribute__((ext_vector_type(8)));
typedef unsigned int u32x4 __attribute__((ext_vector_type(4)));
typedef int   i32x4 __attribute__((ext_vector_type(4)));
typedef int   i32x8 __attribute__((ext_vector_type(8)));

#define HW     224
#define PLANE  (HW*HW)          // 50176
#define WPROW  56               // windows per image row
#define WGS_PER_IMG (WPROW*7)   // 56 window-rows * 7 blocks of 8 windows

// Fused windowed patch attention:
//   TDM tile -> LDS, QKV 1x1 conv (WMMA f32 16x16x4), lane-per-row softmax
//   (no cross-lane shuffles; k/v rows via broadcast ds_load_b128),
//   proj 1x1 conv (WMMA) + bias, scattered stores.
// One workgroup = 8 windows (4 rows x 32 cols x 16 ch tile), one wave/window.
__global__ __launch_bounds__(256) void patch_attn_fused(
    const float* __restrict__ x,     const float* __restrict__ wqkv,
    const float* __restrict__ wproj, const float* __restrict__ bproj,
    const float* __restrict__ relh,  const float* __restrict__ relw,
    float* __restrict__ out)
{
    __shared__ float lds_x[16 * 128];        // [ch][row(4)][col(32)]  from TDM
    __shared__ float lds_qkv[8 * 3 * 256];   // [wave][s][head(16)][pos(16)]
    __shared__ float lds_o[8 * 256];         // [wave][head(16)][pos(16)]
    __shared__ float lds_bias[256];          // [p][q] rel-pos bias table

    const int wg   = blockIdx.x;
    const int b    = wg / WGS_PER_IMG;
    const int rr   = wg % WGS_PER_IMG;
    const int wrow = rr / 7;            // window row   [0,56)
    const int wblk = rr % 7;            // block of 8 windows along W

    const int tid  = threadIdx.x;
    const int wv   = tid >> 5;          // wave id = local window id [0,8)
    const int lane = tid & 31;
    const int half = lane >> 4;         // 16-lane half
    const int n    = lane & 15;         // N column / position / attention row

    // ---------------- Phase 0: TDM x-tile -> LDS, build bias table ---------
    {   // one bias entry per thread: bias[p][q]
        const int p = tid >> 4, q = tid & 15;
        lds_bias[tid] = relh[(p >> 2) - (q >> 2) + 3] + relw[(p & 3) - (q & 3) + 3];
    }
    if (wv == 0) {
        uint64_t ga = (uint64_t)(uintptr_t)x +
            4ull * ((uint64_t)b * 16 * PLANE + (uint64_t)wrow * 4 * HW +
                    (uint64_t)wblk * 32);
        uint32_t ldsa = (uint32_t)(uintptr_t)lds_x;   // flat LDS addr low32 = offset
        u32x4 g0;
        g0[0] = 1u;                                   // count=1, user descriptor
        g0[1] = ldsa;                                 // lds_addr
        g0[2] = (uint32_t)ga;                         // global_addr[31:0]
        g0[3] = ((uint32_t)(ga >> 32) & 0x01FFFFFFu) | (2u << 30);  // type=2
        i32x8 g1;
        g1[0] = (int)(2u << 16);                      // data_size = 4 bytes
        g1[1] = (int)(224u << 16);                    // tensor_dim0 = 224 (lo16)
        g1[2] = (int)(224u << 16);                    // dim0 hi=0 | tensor_dim1=224 lo
        g1[3] = (int)(32u  << 16);                    // dim1 hi=0 | tile_dim0 = 32
        g1[4] = (int)(4u | (16u << 16));              // tile_dim1 = 4, tile_dim2 = 16
        g1[5] = 224;                                  // tensor_dim0_stride = 224
        g1[6] = (int)(50176u << 16);                  // stride0 hi=0 | dim1_stride lo
        g1[7] = 0;                                    // dim1_stride hi (50176>>16==0)
        i32x4 g2; g2[0] = 16; g2[1] = 0; g2[2] = 0; g2[3] = 0;  // tensor_dim2 = 16
        i32x4 g3; g3[0] = 0;  g3[1] = 0; g3[2] = 0; g3[3] = 0;
        i32x8 g4; g4[0] = 0; g4[1] = 0; g4[2] = 0; g4[3] = 0;
                  g4[4] = 0; g4[5] = 0; g4[6] = 0; g4[7] = 0;
        __builtin_amdgcn_tensor_load_to_lds(g0, g1, g2, g3, g4, 0);
        __builtin_amdgcn_s_wait_tensorcnt(0);
    }
    __syncthreads();

    // ---------------- Preload A-fragments (weights, L2-resident) -----------
    // A 16x4 layout: lane m = lane&15 holds row m; v0: K=k0+2*half, v1: +1.
    v2f aq[3][4];
#pragma unroll
    for (int mt = 0; mt < 3; ++mt)
#pragma unroll
        for (int kk = 0; kk < 4; ++kk) {
            const int c0 = 4 * kk + 2 * half;
            aq[mt][kk].x = wqkv[(mt * 16 + n) * 16 + c0];
            aq[mt][kk].y = wqkv[(mt * 16 + n) * 16 + c0 + 1];
        }
    v2f ap[4];
#pragma unroll
    for (int kk = 0; kk < 4; ++kk) {
        const int c0 = 4 * kk + 2 * half;
        ap[kk].x = wproj[n * 16 + c0];
        ap[kk].y = wproj[n * 16 + c0 + 1];
    }

    // ---------------- Phase 1: QKV GEMM  (W_qkv 48x16 @ X 16x16) -----------
    v8f acc[3] = { {0,0,0,0,0,0,0,0}, {0,0,0,0,0,0,0,0}, {0,0,0,0,0,0,0,0} };
    const int bidx = (n >> 2) * 32 + wv * 4 + (n & 3);   // pixel within tile
#pragma unroll
    for (int kk = 0; kk < 4; ++kk) {
        const int c0 = 4 * kk + 2 * half;
        v2f bf;
        bf.x = lds_x[c0 * 128 + bidx];
        bf.y = lds_x[(c0 + 1) * 128 + bidx];
#pragma unroll
        for (int mt = 0; mt < 3; ++mt)
            acc[mt] = __builtin_amdgcn_wmma_f32_16x16x4_f32(
                false, aq[mt][kk], false, bf, (short)0, acc[mt], false, false);
    }
    // Spill Q/K/V to LDS in [wave][s][head][pos] order (lane=pos: conflict-free).
    // Accumulator layout: VGPR r = heads r (lanes 0-15) / r+8 (lanes 16-31).
#pragma unroll
    for (int s = 0; s < 3; ++s)
#pragma unroll
        for (int r = 0; r < 8; ++r)
            lds_qkv[((wv * 3 + s) * 16 + (r + 8 * half)) * 16 + n] = acc[s][r];
    __syncthreads();

    // ---------------- Phase 2: attention (HD==1, scale==1) -----------------
    // Lane (half, n) owns row p=n of head h=2*pass+half: the whole softmax
    // row is reduced inside the lane -> zero cross-lane shuffles.
    float biasr[16];
    {   // this lane's bias row [p=n][q=0..15], 4x ds_load_b128
        const float4* brow = (const float4*)&lds_bias[n * 16];
#pragma unroll
        for (int i = 0; i < 4; ++i) {
            float4 t = brow[i];
            biasr[4*i+0] = t.x; biasr[4*i+1] = t.y;
            biasr[4*i+2] = t.z; biasr[4*i+3] = t.w;
        }
    }
    const float* qs = &lds_qkv[(wv * 3 + 0) * 256];
    const float* ks = &lds_qkv[(wv * 3 + 1) * 256];
    const float* vs = &lds_qkv[(wv * 3 + 2) * 256];
#pragma unroll
    for (int pp = 0; pp < 8; ++pp) {
        const int h = 2 * pp + half;
        const float qp = qs[h * 16 + n];
        float kr[16], vr[16];
        {   // k/v rows of this head: broadcast b128 loads within each half
            const float4* k4 = (const float4*)&ks[h * 16];
            const float4* v4 = (const float4*)&vs[h * 16];
#pragma unroll
            for (int i = 0; i < 4; ++i) {
                float4 tk = k4[i], tv = v4[i];
                kr[4*i+0] = tk.x; kr[4*i+1] = tk.y; kr[4*i+2] = tk.z; kr[4*i+3] = tk.w;
                vr[4*i+0] = tv.x; vr[4*i+1] = tv.y; vr[4*i+2] = tv.z; vr[4*i+3] = tv.w;
            }
        }
        float a[16];
        float m = -3.0e38f;
#pragma unroll
        for (int q = 0; q < 16; ++q) {
            a[q] = fmaf(qp, kr[q], biasr[q]);
            m = fmaxf(m, a[q]);
        }
        float ssum = 0.0f, num = 0.0f;
#pragma unroll
        for (int q = 0; q < 16; ++q) {
            const float e = __expf(a[q] - m);
            ssum += e;
            num = fmaf(e, vr[q], num);
        }
        lds_o[wv * 256 + h * 16 + n] = num / ssum;
    }
    __syncthreads();

    // ---------------- Phase 3: proj GEMM (W_proj 16x16 @ O 16x16) + bias ---
    v8f accP = {0,0,0,0,0,0,0,0};
#pragma unroll
    for (int kk = 0; kk < 4; ++kk) {
        const int c0 = 4 * kk + 2 * half;
        v2f bf;
        bf.x = lds_o[wv * 256 + c0 * 16 + n];
        bf.y = lds_o[wv * 256 + (c0 + 1) * 16 + n];
        accP = __builtin_amdgcn_wmma_f32_16x16x4_f32(
            false, ap[kk], false, bf, (short)0, accP, false, false);
    }

    const int row = wrow * 4 + (n >> 2);
    const int col = (wblk * 8 + wv) * 4 + (n & 3);
#pragma unroll
    for (int r = 0; r < 8; ++r) {
        const int ch = r + 8 * half;
        out[((b * 16 + ch) * HW + row) * HW + col] = accP[r] + bproj[ch];
    }
}

extern "C" void kernel_launch(void* const* d_in, const int* in_sizes, int n_in,
                              void* d_out, int out_size, void* d_ws, size_t ws_size,
                              hipStream_t stream) {
    const float* x     = (const float*)d_in[0];
    const float* wqkv  = (const float*)d_in[1];
    const float* wproj = (const float*)d_in[2];
    const float* bproj = (const float*)d_in[3];
    const float* relh  = (const float*)d_in[4];
    const float* relw  = (const float*)d_in[5];
    float* out = (float*)d_out;

    dim3 grid(8 * WGS_PER_IMG);   // 8 images * 56 window-rows * 7 blocks = 3136
    dim3 block(256);              // 8 wave32s, one window each
    patch_attn_fused<<<grid, block, 0, stream>>>(x, wqkv, wproj, bproj, relh, relw, out);
}